// GATBackbone_68676527063442
// MI455X (gfx1250) — compile-verified
//
#include <hip/hip_runtime.h>
#include <hip/hip_bf16.h>
#include <math.h>
#include <stdint.h>

typedef __attribute__((ext_vector_type(2))) float v2f;
typedef __attribute__((ext_vector_type(8))) float v8f;

#define NEG_SLOPE 0.2f
#define EPSV 1e-16f

// ---- order-preserving float<->uint encoding for atomicMax on f32 ----
__device__ __forceinline__ unsigned f2ord(float f) {
  unsigned u = __float_as_uint(f);
  return (u & 0x80000000u) ? ~u : (u | 0x80000000u);
}
__device__ __forceinline__ float ord2f(unsigned u) {
  return (u & 0x80000000u) ? __uint_as_float(u & 0x7fffffffu) : __uint_as_float(~u);
}

// ------------------------------------------------------------------
// WMMA GEMM: C[N,NCOLS] = A[N,128] @ B[128,NCOLS], f32.
// One block per 16-row strip; NCOLS/16 waves, one 16x16 tile per wave.
// A strip (16x128, 8KB) is async-staged into LDS once per block via
// GLOBAL_LOAD_ASYNC_TO_LDS_B128 (ASYNCcnt), then WMMA A-frags come
// from LDS (ds_load_b64) while B-frags stream from global/L2.
// ------------------------------------------------------------------
template <int NCOLS>
__global__ void gat_gemm_k128(const float* __restrict__ A,
                              const float* __restrict__ B,
                              float* __restrict__ C, int Nrows) {
  constexpr int WAVES = NCOLS / 16;
  __shared__ float As[16 * 128];

  const int tm = blockIdx.x;
  if (tm * 16 >= Nrows) return;  // block-uniform: EXEC stays all-ones

  const int t = threadIdx.x;
  // stage A strip: each lane copies 16B global -> LDS asynchronously
  for (int i = t * 4; i < 16 * 128; i += WAVES * 32 * 4) {
    const float* g = A + (size_t)(tm * 16 + (i >> 7)) * 128 + (i & 127);
    unsigned l = (unsigned)(uintptr_t)&As[i];
    asm volatile("global_load_async_to_lds_b128 %0, %1, off"
                 :: "v"(l), "v"(g) : "memory");
  }
  asm volatile("s_wait_asynccnt 0x0" ::: "memory");
  __syncthreads();

  const int wv   = t >> 5;            // tile column index within strip
  const int lane = t & 31;
  const int n    = wv * 16 + (lane & 15);
  const int koff = (lane >> 4) * 2;   // half-wave K split per ISA layout
  const float* arow = &As[(lane & 15) * 128];

  v8f acc = {};
#pragma unroll 4
  for (int k0 = 0; k0 < 128; k0 += 4) {
    v2f a, b;
    a.x = arow[k0 + koff];
    a.y = arow[k0 + koff + 1];
    b.x = B[(size_t)(k0 + koff) * NCOLS + n];
    b.y = B[(size_t)(k0 + koff + 1) * NCOLS + n];
    acc = __builtin_amdgcn_wmma_f32_16x16x4_f32(false, a, false, b,
                                                (short)0, acc, false, false);
  }
  // D layout: VGPR r -> row (tm*16 + r + 8*(lane>=16)), col wv*16+(lane&15)
  const int rbase = tm * 16 + ((lane >> 4) << 3);
  const int col   = wv * 16 + (lane & 15);
#pragma unroll
  for (int r = 0; r < 8; ++r)
    C[(size_t)(rbase + r) * NCOLS + col] = acc[r];
}

// ------------------------------------------------------------------
// per-(node,head): a_src/a_dst dot products; init segment max/sum
// ------------------------------------------------------------------
__global__ void gat_attn_scores(const float* __restrict__ h,
                                const float* __restrict__ att_src,
                                const float* __restrict__ att_dst,
                                float* __restrict__ asrc, float* __restrict__ adst,
                                unsigned* __restrict__ mOrd, float* __restrict__ ssum,
                                int N, int H, int C) {
  int i = blockIdx.x * blockDim.x + threadIdx.x;
  if (i >= N * H) return;
  int hd = i % H;
  const float* row = h + (size_t)(i / H) * H * C + (size_t)hd * C;
  const float* ws = att_src + (size_t)hd * C;
  const float* wd = att_dst + (size_t)hd * C;
  float s = 0.f, d = 0.f;
  for (int c = 0; c < C; ++c) { s += row[c] * ws[c]; d += row[c] * wd[c]; }
  asrc[i] = s; adst[i] = d;
  mOrd[i] = 0u;      // ordered-encoding identity (below any real float)
  ssum[i] = 0.f;
}

// ------------------------------------------------------------------
// per-(edge,head): leaky-relu logit, atomic segment max (ordered u32)
// edges [0,E) from edge_index, [E,E+N) are self loops
// ------------------------------------------------------------------
__global__ void gat_edge_max(const long long* __restrict__ ei, long long E, long long Etot,
                             int H, const float* __restrict__ asrc,
                             const float* __restrict__ adst, unsigned* __restrict__ mOrd) {
  long long idx = blockIdx.x * (long long)blockDim.x + threadIdx.x;
  if (idx >= Etot * H) return;
  long long e = idx / H; int h = (int)(idx % H);
  int s = (e < E) ? (int)ei[e]     : (int)(e - E);
  int d = (e < E) ? (int)ei[E + e] : (int)(e - E);
  float v = asrc[(size_t)s * H + h] + adst[(size_t)d * H + h];
  v = (v > 0.f) ? v : NEG_SLOPE * v;
  atomicMax(&mOrd[(size_t)d * H + h], f2ord(v));
}

// ------------------------------------------------------------------
// per-(edge,head): ex = exp(e - m[dst]); cache ex; atomic segment sum
// ------------------------------------------------------------------
__global__ void gat_edge_expsum(const long long* __restrict__ ei, long long E, long long Etot,
                                int H, const float* __restrict__ asrc,
                                const float* __restrict__ adst,
                                const unsigned* __restrict__ mOrd,
                                float* __restrict__ exb, float* __restrict__ ssum) {
  long long idx = blockIdx.x * (long long)blockDim.x + threadIdx.x;
  if (idx >= Etot * H) return;
  long long e = idx / H; int h = (int)(idx % H);
  int s = (e < E) ? (int)ei[e]     : (int)(e - E);
  int d = (e < E) ? (int)ei[E + e] : (int)(e - E);
  float v = asrc[(size_t)s * H + h] + adst[(size_t)d * H + h];
  v = (v > 0.f) ? v : NEG_SLOPE * v;
  float ex = __expf(v - ord2f(mOrd[(size_t)d * H + h]));
  exb[idx] = ex;
  atomicAdd(&ssum[(size_t)d * H + h], ex);
}

// ------------------------------------------------------------------
// wave-per-edge message aggregation: out[dst] += h[src] * alpha
// lane covers channels c = lane, lane+32, ... over H*C contiguous floats
// ------------------------------------------------------------------
__global__ void gat_edge_aggregate(const long long* __restrict__ ei, long long E, long long Etot,
                                   int H, int C, const float* __restrict__ exb,
                                   const float* __restrict__ ssum,
                                   const float* __restrict__ feat,
                                   float* __restrict__ out) {
  long long wave = (blockIdx.x * (long long)blockDim.x + threadIdx.x) >> 5;
  int lane = threadIdx.x & 31;
  if (wave >= Etot) return;   // wave-uniform
  int s = (wave < E) ? (int)ei[wave]     : (int)(wave - E);
  int d = (wave < E) ? (int)ei[E + wave] : (int)(wave - E);
  const int HC = H * C;
  const float* frow = feat + (size_t)s * HC;
  float* orow = out + (size_t)d * HC;
  for (int c = lane; c < HC; c += 32) {
    int h = c / C;
    float alpha = exb[wave * H + h] / (ssum[(size_t)d * H + h] + EPSV);
    atomicAdd(&orow[c], frow[c] * alpha);
  }
}

// ------------------------------------------------------------------
__global__ void gat_bias_elu(float* __restrict__ a, const float* __restrict__ bias,
                             long long total, int F) {
  long long i = blockIdx.x * (long long)blockDim.x + threadIdx.x;
  if (i >= total) return;
  float v = a[i] + bias[i % F];
  a[i] = (v > 0.f) ? v : (__expf(v) - 1.f);
}

__global__ void gat_bias_out(const float* __restrict__ a, const float* __restrict__ bias,
                             float* __restrict__ out, long long total, int F) {
  long long i = blockIdx.x * (long long)blockDim.x + threadIdx.x;
  if (i >= total) return;
  out[i] = a[i] + bias[i % F];
}

__global__ void gat_zero(float* __restrict__ p, long long total) {
  long long i = blockIdx.x * (long long)blockDim.x + threadIdx.x;
  if (i < total) p[i] = 0.f;
}

// ------------------------------------------------------------------
extern "C" void kernel_launch(void* const* d_in, const int* in_sizes, int n_in,
                              void* d_out, int out_size, void* d_ws, size_t ws_size,
                              hipStream_t stream) {
  const float*     x   = (const float*)d_in[0];
  const long long* ei  = (const long long*)d_in[1];
  const float*     W1  = (const float*)d_in[2];
  const float*     as1 = (const float*)d_in[3];
  const float*     ad1 = (const float*)d_in[4];
  const float*     b1  = (const float*)d_in[5];
  const float*     W2  = (const float*)d_in[6];
  const float*     as2 = (const float*)d_in[7];
  const float*     ad2 = (const float*)d_in[8];
  const float*     b2  = (const float*)d_in[9];
  float* out = (float*)d_out;

  const long long N    = in_sizes[0] / 128;
  const long long E    = in_sizes[1] / 2;
  const long long Etot = E + N;

  // workspace carve-up (floats)
  float* ws = (float*)d_ws;
  size_t o = 0;
  float*    h1    = ws + o; o += (size_t)N * 128;   // reused as h2 after layer 1
  float*    agg1  = ws + o; o += (size_t)N * 128;
  float*    asrc1 = ws + o; o += (size_t)N * 2;
  float*    adst1 = ws + o; o += (size_t)N * 2;
  unsigned* m1    = (unsigned*)(ws + o); o += (size_t)N * 2;
  float*    s1    = ws + o; o += (size_t)N * 2;
  float*    asrc2 = ws + o; o += (size_t)N;
  float*    adst2 = ws + o; o += (size_t)N;
  unsigned* m2    = (unsigned*)(ws + o); o += (size_t)N;
  float*    s2    = ws + o; o += (size_t)N;
  float*    agg2  = ws + o; o += (size_t)N * 64;
  float*    exb   = ws + o; o += (size_t)Etot * 2;  // reused (H=1) in layer 2

  const int T = 256;
  auto cdiv = [](long long a, long long b) -> unsigned { return (unsigned)((a + b - 1) / b); };

  // init accumulators
  gat_zero<<<cdiv(N * 128, T), T, 0, stream>>>(agg1, N * 128);
  gat_zero<<<cdiv(N * 64, T), T, 0, stream>>>(agg2, N * 64);

  // ---------------- layer 1 (H=2, C=64, concat) ----------------
  gat_gemm_k128<128><<<cdiv(N, 16), 256, 0, stream>>>(x, W1, h1, (int)N);
  gat_attn_scores<<<cdiv(N * 2, T), T, 0, stream>>>(h1, as1, ad1, asrc1, adst1, m1, s1,
                                                    (int)N, 2, 64);
  gat_edge_max<<<cdiv(Etot * 2, T), T, 0, stream>>>(ei, E, Etot, 2, asrc1, adst1, m1);
  gat_edge_expsum<<<cdiv(Etot * 2, T), T, 0, stream>>>(ei, E, Etot, 2, asrc1, adst1, m1,
                                                       exb, s1);
  gat_edge_aggregate<<<cdiv(Etot * 32, T), T, 0, stream>>>(ei, E, Etot, 2, 64, exb, s1,
                                                           h1, agg1);
  gat_bias_elu<<<cdiv(N * 128, T), T, 0, stream>>>(agg1, b1, N * 128, 128);

  // ---------------- layer 2 (H=1, C=64, mean==identity) ----------------
  float* h2 = h1;  // reuse
  gat_gemm_k128<64><<<cdiv(N, 16), 128, 0, stream>>>(agg1, W2, h2, (int)N);
  gat_attn_scores<<<cdiv(N, T), T, 0, stream>>>(h2, as2, ad2, asrc2, adst2, m2, s2,
                                                (int)N, 1, 64);
  gat_edge_max<<<cdiv(Etot, T), T, 0, stream>>>(ei, E, Etot, 1, asrc2, adst2, m2);
  gat_edge_expsum<<<cdiv(Etot, T), T, 0, stream>>>(ei, E, Etot, 1, asrc2, adst2, m2,
                                                   exb, s2);
  gat_edge_aggregate<<<cdiv(Etot * 32, T), T, 0, stream>>>(ei, E, Etot, 1, 64, exb, s2,
                                                           h2, agg2);
  gat_bias_out<<<cdiv(N * 64, T), T, 0, stream>>>(agg2, b2, out, N * 64, 64);
}